// Qwen3NextGatedDeltaNet_15083925143802
// MI455X (gfx1250) — compile-verified
//
#include <hip/hip_runtime.h>
#include <hip/hip_bf16.h>

typedef float v2f __attribute__((ext_vector_type(2)));
typedef float v8f __attribute__((ext_vector_type(8)));

#define T_LEN   2048
#define HID     2048
#define NUM_K   16
#define NUM_V   32
#define DK      128
#define DV      128
#define KEY_DIM 2048
#define VAL_DIM 4096
#define QKVZ_N  12288          // 2*KEY + 2*VAL
#define CONV_CH 8192           // q+k+v

// ---------------------------------------------------------------------------
// CDNA5 async memory->LDS copy (ASYNCcnt-tracked).  Inline asm is the
// toolchain-portable spelling (builtin arity differs between ROCm7.2 and
// amdgpu-toolchain).  GVS mode: 64-bit SGPR base + 32-bit unsigned VGPR
// byte offset; VDST VGPR holds the wave-relative LDS byte address.
// ---------------------------------------------------------------------------
__device__ __forceinline__ void async_b128(unsigned lds_byte,
                                           const float* __restrict__ base,
                                           unsigned voff_byte)
{
    asm volatile("global_load_async_to_lds_b128 %0, %1, %2"
                 :: "v"(lds_byte), "v"(voff_byte), "s"(base)
                 : "memory");
}
#define WAIT_ASYNC_0() asm volatile("s_wait_asynccnt 0x0" ::: "memory")
#define WAIT_ASYNC_1() asm volatile("s_wait_asynccnt 0x1" ::: "memory")
#define WAIT_ASYNC_8() asm volatile("s_wait_asynccnt 0x8" ::: "memory")

// ---------------------------------------------------------------------------
// WMMA f32 GEMM: C[M,N] = A[M,K] * B[K,N], row-major, M%128==0, N%128==0,
// K%32==0.  Block tile 128x128, 8 waves of 32(M)x64(N) -> 2x4 v8f accs.
// K staged in 32-wide LDS slabs, DOUBLE-BUFFERED via async-to-LDS copies:
// issue slab k+1, compute WMMA on slab k, s_wait_asynccnt + barrier.
// Full f32 via V_WMMA_F32_16X16X4_F32 (workload is HBM-bound at 23.3TB/s,
// so no precision downgrade is needed to reach the roofline).
// ---------------------------------------------------------------------------
#define AS_LD 36   // 36 floats = 144B rows: 16B-aligned, 16-lane stride hits 16 banks
#define BS_LD 132  // 132 floats = 528B rows: 16B-aligned, conflict-spread

__global__ __launch_bounds__(256) void gemm_f32_wmma(
    const float* __restrict__ A, const float* __restrict__ B,
    float* __restrict__ C, int M, int N, int K)
{
    __shared__ float As[2][128 * AS_LD];
    __shared__ float Bs[2][32 * BS_LD];

    const int tid  = threadIdx.x;
    const int lane = tid & 31;
    const int wave = tid >> 5;
    const int wm0  = (wave & 3) * 32;   // wave row offset in block tile
    const int wn0  = (wave >> 2) * 64;  // wave col offset in block tile
    const int m0   = blockIdx.y * 128;
    const int n0   = blockIdx.x * 128;
    const int half = lane >> 4;         // selects K pair {0,1} vs {2,3}
    const int l15  = lane & 15;

    const unsigned asb = (unsigned)(uintptr_t)(&As[0][0]);
    const unsigned bsb = (unsigned)(uintptr_t)(&Bs[0][0]);

    // stage one 128x32 A slab + 32x128 B slab into LDS buffer `buf`
    // (8 async b128 per thread per stage)
    auto stage = [&](int buf, int k0) {
#pragma unroll
        for (int i = 0; i < 4; ++i) {
            const int row = (tid >> 3) + i * 32;
            const int col = (tid & 7) * 4;
            async_b128(asb + (unsigned)((buf * 128 * AS_LD + row * AS_LD + col) * 4),
                       A, (unsigned)(((size_t)(m0 + row) * K + k0 + col) * 4));
            const int krow = (tid >> 5) + i * 8;
            const int bcol = (tid & 31) * 4;
            async_b128(bsb + (unsigned)((buf * 32 * BS_LD + krow * BS_LD + bcol) * 4),
                       B, (unsigned)(((size_t)(k0 + krow) * N + n0 + bcol) * 4));
        }
    };

    v8f acc[2][4];
#pragma unroll
    for (int i = 0; i < 2; ++i)
#pragma unroll
        for (int j = 0; j < 4; ++j) acc[i][j] = (v8f)0.0f;

    stage(0, 0);                         // prologue

    for (int k0 = 0; k0 < K; k0 += 32) {
        const int cur = (k0 >> 5) & 1;
        if (k0 + 32 < K) {               // overlap next slab's HBM fetch
            stage(cur ^ 1, k0 + 32);
            WAIT_ASYNC_8();              // 8 just issued; wait current slab done
        } else {
            WAIT_ASYNC_0();
        }
        __syncthreads();                 // slab `cur` visible to all waves

        const float* as = &As[cur][0];
        const float* bs = &Bs[cur][0];
#pragma unroll
        for (int kk = 0; kk < 32; kk += 4) {
            v2f af[2], bf[4];
            // A frag 16x4: lanes 0-15 hold K=kk,kk+1; lanes 16-31 K=kk+2,kk+3
#pragma unroll
            for (int ti = 0; ti < 2; ++ti) {
                const int r = wm0 + ti * 16 + l15;
                af[ti].x = as[r * AS_LD + kk + 2 * half];
                af[ti].y = as[r * AS_LD + kk + 2 * half + 1];
            }
            // B frag 4x16: rows striped across lanes within a VGPR
#pragma unroll
            for (int tj = 0; tj < 4; ++tj) {
                const int cn = wn0 + tj * 16 + l15;
                bf[tj].x = bs[(kk + 2 * half) * BS_LD + cn];
                bf[tj].y = bs[(kk + 2 * half + 1) * BS_LD + cn];
            }
#pragma unroll
            for (int ti = 0; ti < 2; ++ti)
#pragma unroll
                for (int tj = 0; tj < 4; ++tj)
                    acc[ti][tj] = __builtin_amdgcn_wmma_f32_16x16x4_f32(
                        false, af[ti], false, bf[tj], (short)0, acc[ti][tj],
                        false, false);
        }
        __syncthreads();                 // reads done before `cur` is re-staged
    }

    // C/D layout: VGPR r -> (M=r, N=lane) lanes 0-15, (M=8+r, N=lane-16) 16-31
#pragma unroll
    for (int ti = 0; ti < 2; ++ti)
#pragma unroll
        for (int tj = 0; tj < 4; ++tj)
#pragma unroll
            for (int r = 0; r < 8; ++r) {
                const int mr = m0 + wm0 + ti * 16 + half * 8 + r;
                const int nc = n0 + wn0 + tj * 16 + l15;
                C[(size_t)mr * N + nc] = acc[ti][tj][r];
            }
}

// ---------------------------------------------------------------------------
// Tiny ba GEMM: [T x HID] * [HID x 64]  (0.27 GFLOP, scalar is fine)
// ---------------------------------------------------------------------------
__global__ __launch_bounds__(256) void gemm_ba(
    const float* __restrict__ hs, const float* __restrict__ Wba,
    float* __restrict__ out)
{
    const int idx = blockIdx.x * blockDim.x + threadIdx.x;  // T*64
    const int t = idx >> 6, j = idx & 63;
    float acc = 0.f;
    for (int k = 0; k < HID; ++k)
        acc = fmaf(hs[(size_t)t * HID + k], Wba[(size_t)k * 64 + j], acc);
    out[idx] = acc;
}

// ---------------------------------------------------------------------------
// Causal depthwise conv(K=4) + SiLU over [q|k|v] (cols 0..8191 of qkvz),
// per-head L2-norm of q and k, plus g = -exp(A_log)*softplus(a+dt_bias),
// beta = sigmoid(b).  One block per token; thread tid owns 32 contiguous
// channels (4 threads per 128-wide head).
// ---------------------------------------------------------------------------
__global__ __launch_bounds__(256) void conv_gate(
    const float* __restrict__ qkvz, const float* __restrict__ ba,
    const float* __restrict__ conv_w, const float* __restrict__ dt_bias,
    const float* __restrict__ A_log,
    float* __restrict__ qpost, float* __restrict__ kpost,
    float* __restrict__ vpost, float* __restrict__ g, float* __restrict__ beta)
{
    const int t = blockIdx.x;
    const int tid = threadIdx.x;
    const int c0 = tid * 32;
    __shared__ float red[128];

    float yv[32];
    float sumsq = 0.f;
#pragma unroll
    for (int i = 0; i < 32; ++i) {
        const int c = c0 + i;
        float acc = 0.f;
#pragma unroll
        for (int j = 0; j < 4; ++j) {
            const int tt = t - 3 + j;
            if (tt >= 0)
                acc = fmaf(qkvz[(size_t)tt * QKVZ_N + c], conv_w[c * 4 + j], acc);
        }
        const float s = acc / (1.f + __expf(-acc));  // SiLU
        yv[i] = s;
        sumsq += s * s;
    }

    if (tid < 128) red[tid] = sumsq;   // q/k region partials
    __syncthreads();

    if (tid < 128) {
        const int base = tid & ~3;     // 4 threads per 128-ch head
        const float rn = rsqrtf(red[base] + red[base + 1] + red[base + 2] +
                                red[base + 3] + 1e-6f);
#pragma unroll
        for (int i = 0; i < 32; ++i) {
            const int c = c0 + i;
            const float v = yv[i] * rn;
            if (c < KEY_DIM) qpost[(size_t)t * KEY_DIM + c] = v;
            else             kpost[(size_t)t * KEY_DIM + (c - KEY_DIM)] = v;
        }
    } else {
#pragma unroll
        for (int i = 0; i < 32; ++i) {
            const int c = c0 + i;      // >= 4096 -> v region
            vpost[(size_t)t * VAL_DIM + (c - 2 * KEY_DIM)] = yv[i];
        }
    }

    if (tid < NUM_V) {
        const float bv = ba[t * 2 * NUM_V + tid];
        const float av = ba[t * 2 * NUM_V + NUM_V + tid];
        const float x  = av + dt_bias[tid];
        const float sp = (x > 20.f) ? x : log1pf(__expf(x));   // softplus
        g[t * NUM_V + tid]    = -__expf(A_log[tid]) * sp;
        beta[t * NUM_V + tid] = 1.f / (1.f + __expf(-bv));
    }
}

// ---------------------------------------------------------------------------
// Gated delta-rule scan: one block per v-head (32 blocks), state S[128][128]
// f32 lives in registers (64 floats/thread).  thread -> (dv = tid&127,
// dk segment = (tid>>7)*64).  Per-step k/q/v (384 floats) are DOUBLE-BUFFERED
// in LDS via async-to-LDS: waves 0/1/2 each issue one b128 per lane for
// step t+1 while step t computes, removing HBM latency from the recurrence
// critical path.  GQA: key head = h >> 1.
// ---------------------------------------------------------------------------
__global__ __launch_bounds__(256) void gdn_scan(
    const float* __restrict__ qpost, const float* __restrict__ kpost,
    const float* __restrict__ vpost, const float* __restrict__ g,
    const float* __restrict__ beta, float* __restrict__ o)
{
    const int h   = blockIdx.x;
    const int kh  = h >> 1;
    const int tid = threadIdx.x;
    const int dv  = tid & 127;
    const int seg = tid >> 7;
    const int dk0 = seg * 64;
    const int wv  = tid >> 5;
    const int ln  = tid & 31;

    __shared__ float kqv[2][384];       // [k:0..127 | q:128..255 | v:256..383]
    __shared__ float red[256];

    const unsigned kqvb = (unsigned)(uintptr_t)(&kqv[0][0]);

    auto issue = [&](int tt, int buf) { // one b128 per lane, waves 0..2
        const unsigned ldsb = kqvb + (unsigned)(buf * 384 * 4);
        if (wv == 0)
            async_b128(ldsb + (unsigned)(ln * 16), kpost,
                       (unsigned)(((size_t)tt * KEY_DIM + kh * DK) * 4 + ln * 16));
        else if (wv == 1)
            async_b128(ldsb + 512u + (unsigned)(ln * 16), qpost,
                       (unsigned)(((size_t)tt * KEY_DIM + kh * DK) * 4 + ln * 16));
        else if (wv == 2)
            async_b128(ldsb + 1024u + (unsigned)(ln * 16), vpost,
                       (unsigned)(((size_t)tt * VAL_DIM + h * DV) * 4 + ln * 16));
    };

    float S[64];
#pragma unroll
    for (int i = 0; i < 64; ++i) S[i] = 0.f;
    const float scale = 0.08838834764831845f;  // 128^-0.5

    issue(0, 0);                        // prologue

    for (int t = 0; t < T_LEN; ++t) {
        const int cur = t & 1;
        if (t + 1 < T_LEN) {
            issue(t + 1, cur ^ 1);      // prefetch next step
            WAIT_ASYNC_1();             // wait for step t's buffer
        } else {
            WAIT_ASYNC_0();
        }
        const float eg = __expf(g[t * NUM_V + h]);
        const float bt = beta[t * NUM_V + h];
        __syncthreads();                                    // sync1

        const float* kb = &kqv[cur][0];
        const float* qb = &kqv[cur][128];
        const float* vb = &kqv[cur][256];

        // S *= exp(g); kv = k . S   (partial over this thread's dk segment)
        float pkv = 0.f;
#pragma unroll 8
        for (int i = 0; i < 64; ++i) {
            const float s = S[i] * eg;
            S[i] = s;
            pkv = fmaf(kb[dk0 + i], s, pkv);
        }
        red[tid] = pkv;
        __syncthreads();                                    // sync2

        const float delta = (vb[dv] - (red[dv] + red[dv + 128])) * bt;
        __syncthreads();   // all delta reads done before red rewrite  sync3

        // S += k (x) delta ; o = (q*scale) . S
        float po = 0.f;
#pragma unroll 8
        for (int i = 0; i < 64; ++i) {
            const float s = fmaf(kb[dk0 + i], delta, S[i]);
            S[i] = s;
            po = fmaf(qb[dk0 + i], s, po);
        }
        red[tid] = po;
        __syncthreads();                                    // sync4

        if (seg == 0)
            o[(size_t)t * VAL_DIM + h * DV + dv] = (red[dv] + red[dv + 128]) * scale;
        // buffer `cur` is re-staged only after every wave passes sync4
    }
}

// ---------------------------------------------------------------------------
// Gated RMSNorm (norm before gate) * SiLU(z):  one block per (head, token)
// ---------------------------------------------------------------------------
__global__ __launch_bounds__(128) void norm_gate(
    const float* __restrict__ o, const float* __restrict__ qkvz,
    const float* __restrict__ norm_w, float* __restrict__ gated)
{
    const int h  = blockIdx.x;
    const int t  = blockIdx.y;
    const int dv = threadIdx.x;
    __shared__ float red[128];

    const float val = o[(size_t)t * VAL_DIM + h * DV + dv];
    red[dv] = val * val;
    __syncthreads();
#pragma unroll
    for (int s = 64; s > 0; s >>= 1) {
        if (dv < s) red[dv] += red[dv + s];
        __syncthreads();
    }
    const float mean = red[0] * (1.f / 128.f);
    const float z = qkvz[(size_t)t * QKVZ_N + 2 * KEY_DIM + VAL_DIM + h * DV + dv];
    const float out = val * rsqrtf(mean + 1e-6f) * norm_w[dv] *
                      (z / (1.f + __expf(-z)));
    gated[(size_t)t * VAL_DIM + h * DV + dv] = out;
}

// ---------------------------------------------------------------------------
extern "C" void kernel_launch(void* const* d_in, const int* in_sizes, int n_in,
                              void* d_out, int out_size, void* d_ws, size_t ws_size,
                              hipStream_t stream)
{
    const float* hs      = (const float*)d_in[0];
    const float* W_qkvz  = (const float*)d_in[1];
    const float* W_ba    = (const float*)d_in[2];
    const float* conv_w  = (const float*)d_in[3];
    const float* dt_bias = (const float*)d_in[4];
    const float* A_log   = (const float*)d_in[5];
    const float* norm_w  = (const float*)d_in[6];
    const float* W_out   = (const float*)d_in[7];
    float*       out     = (float*)d_out;
    float*       ws      = (float*)d_ws;

    // workspace layout (floats)
    float* qkvz  = ws;                                   // T*12288
    float* ba    = qkvz  + (size_t)T_LEN * QKVZ_N;       // T*64
    float* qpost = ba    + (size_t)T_LEN * 2 * NUM_V;    // T*2048
    float* kpost = qpost + (size_t)T_LEN * KEY_DIM;      // T*2048
    float* vpost = kpost + (size_t)T_LEN * KEY_DIM;      // T*4096
    float* gbuf  = vpost + (size_t)T_LEN * VAL_DIM;      // T*32
    float* bbuf  = gbuf  + (size_t)T_LEN * NUM_V;        // T*32
    float* obuf  = bbuf  + (size_t)T_LEN * NUM_V;        // T*4096
    float* gated = obuf  + (size_t)T_LEN * VAL_DIM;      // T*4096

    // 1) qkvz = hs @ W_qkvz   (WMMA f32, async double-buffered tiles)
    gemm_f32_wmma<<<dim3(QKVZ_N / 128, T_LEN / 128), 256, 0, stream>>>(
        hs, W_qkvz, qkvz, T_LEN, QKVZ_N, HID);
    // 2) ba = hs @ W_ba
    gemm_ba<<<(T_LEN * 2 * NUM_V) / 256, 256, 0, stream>>>(hs, W_ba, ba);
    // 3) conv + SiLU + l2norm + gating
    conv_gate<<<T_LEN, 256, 0, stream>>>(qkvz, ba, conv_w, dt_bias, A_log,
                                         qpost, kpost, vpost, gbuf, bbuf);
    // 4) sequential gated delta-rule scan (state in regs, async prefetch)
    gdn_scan<<<NUM_V, 256, 0, stream>>>(qpost, kpost, vpost, gbuf, bbuf, obuf);
    // 5) RMSNorm * SiLU(z)
    norm_gate<<<dim3(NUM_V, T_LEN), 128, 0, stream>>>(obuf, qkvz, norm_w, gated);
    // 6) out = gated @ W_out  (WMMA f32), straight into d_out
    gemm_f32_wmma<<<dim3(HID / 128, T_LEN / 128), 256, 0, stream>>>(
        gated, W_out, out, T_LEN, HID, VAL_DIM);
}